// GraphEmbedding_49976239456828
// MI455X (gfx1250) — compile-verified
//
#include <hip/hip_runtime.h>
#include <hip/hip_bf16.h>

typedef __attribute__((ext_vector_type(2))) float v2f;
typedef __attribute__((ext_vector_type(8))) float v8f;

#define NGRAPH 64

// ---------------- ordered-float encode for atomicMax on f32 ----------------
__device__ __forceinline__ unsigned int fenc(float f) {
  unsigned int u = __float_as_uint(f);
  return (u & 0x80000000u) ? ~u : (u | 0x80000000u);
}
__device__ __forceinline__ float fdec(unsigned int u) {
  return __uint_as_float((u & 0x80000000u) ? (u & 0x7FFFFFFFu) : ~u);
}

// ---------------- WMMA GEMM: C[M,128] = A[M,K] @ W[K,128] ----------------
// block = 256 threads = 8 wave32; wave w computes cols [16w,16w+16).
// A-tile (16 x K) staged in LDS; K is a multiple of 4 (8/12/16/128 here).
// v_wmma_f32_16x16x4_f32 operand layout (ISA 7.12.2):
//   A 16x4: lanes 0-15 rows, VGPR0=K0 / VGPR1=K1; lanes 16-31: K2/K3
//   B 4x16: row striped across lanes per VGPR (K0/K2 in v0, K1/K3 in v1)
//   C/D: VGPR v -> row v (lanes 0-15) / row v+8 (lanes 16-31)
__global__ void __launch_bounds__(256)
k_gemm_wmma(const float* __restrict__ A, int lda, int K,
            const float* __restrict__ W, float* __restrict__ C, int M) {
  __shared__ float As[16][132];
  const int m0  = blockIdx.x * 16;
  const int tid = threadIdx.x;
  for (int idx = tid; idx < 16 * K; idx += 256) {
    int r = idx / K, c = idx - r * K;
    int gr = m0 + r;
    As[r][c] = (gr < M) ? A[(size_t)gr * lda + c] : 0.0f;
  }
  __syncthreads();
  const int wave = tid >> 5;
  const int lane = tid & 31;
  const int lrow = lane & 15;
  const int hi   = lane >> 4;            // 0 or 1 (half-wave)
  const int col  = wave * 16 + lrow;     // output column 0..127
  v8f acc = {0.f, 0.f, 0.f, 0.f, 0.f, 0.f, 0.f, 0.f};
  for (int k0 = 0; k0 < K; k0 += 4) {
    v2f a, b;
    a.x = As[lrow][k0 + 2 * hi];
    a.y = As[lrow][k0 + 2 * hi + 1];
    b.x = W[(size_t)(k0 + 2 * hi) * 128 + col];
    b.y = W[(size_t)(k0 + 2 * hi + 1) * 128 + col];
    acc = __builtin_amdgcn_wmma_f32_16x16x4_f32(false, a, false, b,
                                                (short)0, acc, false, false);
  }
#pragma unroll
  for (int v = 0; v < 8; ++v) {
    int row = m0 + hi * 8 + v;
    if (row < M) C[(size_t)row * 128 + col] = acc[v];
  }
}

// ---------------- misc small kernels ----------------
__global__ void __launch_bounds__(256)
k_count(const int* __restrict__ b, int* __restrict__ cnt, int N) {
  int i = blockIdx.x * 256 + threadIdx.x;
  if (i < N) atomicAdd(&cnt[b[i]], 1);
}

__global__ void __launch_bounds__(256)
k_bias(float* __restrict__ o, const float* __restrict__ b0,
       const float* __restrict__ b1, const float* __restrict__ b2, int n128) {
  int i = blockIdx.x * 256 + threadIdx.x;
  if (i >= n128) return;
  int c = i & 127;
  float v = b0[c];
  if (b1) v += b1[c];
  if (b2) v += b2[c];
  o[i] = v;
}

// ---------------- GATv2 edge passes ----------------
// pass A: per (edge, head) attention logit + segment max (ordered-uint atomicMax)
// float4 loads -> global_load_b128 on the 64B-aligned per-head slices.
__global__ void __launch_bounds__(256)
k_edge_alpha(const float* __restrict__ xl, const float* __restrict__ xr,
             const int* __restrict__ src, const int* __restrict__ dst,
             const float* __restrict__ eattr, const float* __restrict__ We,
             const float* __restrict__ att,
             float* __restrict__ alpha, unsigned int* __restrict__ amax, int E) {
  int i = blockIdx.x * 256 + threadIdx.x;
  if (i >= E * 8) return;
  int e = i >> 3, h = i & 7;
  int s = src[e], d = dst[e];
  const float4* pl = (const float4*)(xl + (size_t)s * 128 + h * 16);
  const float4* pr = (const float4*)(xr + (size_t)d * 128 + h * 16);
  const float4* pa = (const float4*)(att + h * 16);
  float acc = 0.f;
  if (eattr) {
    float4 ea4 = *(const float4*)(eattr + (size_t)e * 4);
    const float4* pw = (const float4*)(We + h * 16);
#pragma unroll
    for (int j = 0; j < 4; ++j) {
      float4 l4 = pl[j], r4 = pr[j], a4 = pa[j];
      float4 w0 = pw[j], w1 = pw[32 + j], w2 = pw[64 + j], w3 = pw[96 + j];
      float m;
      m = l4.x + r4.x + ea4.x * w0.x + ea4.y * w1.x + ea4.z * w2.x + ea4.w * w3.x;
      acc += a4.x * (m > 0.f ? m : 0.2f * m);
      m = l4.y + r4.y + ea4.x * w0.y + ea4.y * w1.y + ea4.z * w2.y + ea4.w * w3.y;
      acc += a4.y * (m > 0.f ? m : 0.2f * m);
      m = l4.z + r4.z + ea4.x * w0.z + ea4.y * w1.z + ea4.z * w2.z + ea4.w * w3.z;
      acc += a4.z * (m > 0.f ? m : 0.2f * m);
      m = l4.w + r4.w + ea4.x * w0.w + ea4.y * w1.w + ea4.z * w2.w + ea4.w * w3.w;
      acc += a4.w * (m > 0.f ? m : 0.2f * m);
    }
  } else {
#pragma unroll
    for (int j = 0; j < 4; ++j) {
      float4 l4 = pl[j], r4 = pr[j], a4 = pa[j];
      float m;
      m = l4.x + r4.x; acc += a4.x * (m > 0.f ? m : 0.2f * m);
      m = l4.y + r4.y; acc += a4.y * (m > 0.f ? m : 0.2f * m);
      m = l4.z + r4.z; acc += a4.z * (m > 0.f ? m : 0.2f * m);
      m = l4.w + r4.w; acc += a4.w * (m > 0.f ? m : 0.2f * m);
    }
  }
  alpha[i] = acc;
  atomicMax(&amax[d * 8 + h], fenc(acc));
}

// pass B: exp(alpha - max) + segment sum of exps
__global__ void __launch_bounds__(256)
k_edge_exp(float* __restrict__ alpha, const int* __restrict__ dst,
           const unsigned int* __restrict__ amax, float* __restrict__ denom, int E) {
  int i = blockIdx.x * 256 + threadIdx.x;
  if (i >= E * 8) return;
  int e = i >> 3, h = i & 7;
  int d = dst[e];
  float ex = __expf(alpha[i] - fdec(amax[d * 8 + h]));
  alpha[i] = ex;
  atomicAdd(&denom[d * 8 + h], ex);
}

// pass C: out[dst] += xl[src] * a.  One thread per (edge, 4-channel chunk):
// b128 load of xl, 4x fewer index fetches; 4 f32 atomics.
__global__ void __launch_bounds__(256)
k_edge_aggr(const float* __restrict__ xl, const int* __restrict__ src,
            const int* __restrict__ dst, const float* __restrict__ ex,
            const float* __restrict__ denom, float* __restrict__ o, int E) {
  int i = blockIdx.x * 256 + threadIdx.x;
  if (i >= E * 32) return;
  int e = i >> 5, q = i & 31, c = q * 4, h = q >> 2;
  int s = src[e], d = dst[e];
  float a = ex[e * 8 + h] / (denom[d * 8 + h] + 1e-16f);
  float4 v4 = *(const float4*)(xl + (size_t)s * 128 + c);
  float* po = o + (size_t)d * 128 + c;
  atomicAdd(po + 0, v4.x * a);
  atomicAdd(po + 1, v4.y * a);
  atomicAdd(po + 2, v4.z * a);
  atomicAdd(po + 3, v4.w * a);
}

// ---------------- LeakyReLU + GraphNorm (float4 per thread) ----------------
__global__ void __launch_bounds__(256)
k_act_sum(float* __restrict__ x, const int* __restrict__ batch,
          float* __restrict__ mean_acc, int N) {
  int i = blockIdx.x * 256 + threadIdx.x;
  if (i >= N * 32) return;
  int n = i >> 5, c = (i & 31) * 4;
  float4 v = *(float4*)(x + (size_t)n * 128 + c);
  v.x = v.x > 0.f ? v.x : 0.01f * v.x;
  v.y = v.y > 0.f ? v.y : 0.01f * v.y;
  v.z = v.z > 0.f ? v.z : 0.01f * v.z;
  v.w = v.w > 0.f ? v.w : 0.01f * v.w;
  *(float4*)(x + (size_t)n * 128 + c) = v;
  float* pm = mean_acc + batch[n] * 128 + c;
  atomicAdd(pm + 0, v.x);
  atomicAdd(pm + 1, v.y);
  atomicAdd(pm + 2, v.z);
  atomicAdd(pm + 3, v.w);
}

__global__ void __launch_bounds__(256)
k_div_cnt(float* __restrict__ acc, const int* __restrict__ cnt) {
  int i = blockIdx.x * 256 + threadIdx.x;
  if (i >= NGRAPH * 128) return;
  int g = i >> 7;
  float c = (float)(cnt[g] > 1 ? cnt[g] : 1);
  acc[i] = acc[i] / c;
}

__global__ void __launch_bounds__(256)
k_center_var(float* __restrict__ x, const int* __restrict__ batch,
             const float* __restrict__ mean, const float* __restrict__ ms,
             float* __restrict__ var_acc, int N) {
  int i = blockIdx.x * 256 + threadIdx.x;
  if (i >= N * 32) return;
  int n = i >> 5, c = (i & 31) * 4;
  int g = batch[n];
  float4 v  = *(float4*)(x + (size_t)n * 128 + c);
  float4 m4 = *(const float4*)(mean + g * 128 + c);
  float4 s4 = *(const float4*)(ms + c);
  v.x -= m4.x * s4.x;
  v.y -= m4.y * s4.y;
  v.z -= m4.z * s4.z;
  v.w -= m4.w * s4.w;
  *(float4*)(x + (size_t)n * 128 + c) = v;
  float* pv = var_acc + g * 128 + c;
  atomicAdd(pv + 0, v.x * v.x);
  atomicAdd(pv + 1, v.y * v.y);
  atomicAdd(pv + 2, v.z * v.z);
  atomicAdd(pv + 3, v.w * v.w);
}

__global__ void __launch_bounds__(256)
k_var_inv(float* __restrict__ var_acc, const int* __restrict__ cnt) {
  int i = blockIdx.x * 256 + threadIdx.x;
  if (i >= NGRAPH * 128) return;
  int g = i >> 7;
  float c = (float)(cnt[g] > 1 ? cnt[g] : 1);
  var_acc[i] = rsqrtf(var_acc[i] / c + 1e-5f);
}

__global__ void __launch_bounds__(256)
k_norm_final(const float* __restrict__ x, const int* __restrict__ batch,
             const float* __restrict__ inv, const float* __restrict__ w,
             const float* __restrict__ b, float* __restrict__ out, int N) {
  int i = blockIdx.x * 256 + threadIdx.x;
  if (i >= N * 32) return;
  int n = i >> 5, c = (i & 31) * 4;
  int g = batch[n];
  float4 v  = *(const float4*)(x + (size_t)n * 128 + c);
  float4 i4 = *(const float4*)(inv + g * 128 + c);
  float4 w4 = *(const float4*)(w + c);
  float4 b4 = *(const float4*)(b + c);
  float4 y;
  y.x = w4.x * v.x * i4.x + b4.x;
  y.y = w4.y * v.y * i4.y + b4.y;
  y.z = w4.z * v.z * i4.z + b4.z;
  y.w = w4.w * v.w * i4.w + b4.w;
  *(float4*)(out + (size_t)n * 256 + c) = y;   // out pre-offset by type base + layer*128
}

// ---------------- host orchestration ----------------
// param leaf indices: jax tree-flatten (alphabetical): gn{b,ms,w}, l0{bc,bcr,bj,bjr,
// pre,proc,procr}, l1{...}; per-GAT dict order: [We], Wl, Wr, att, b.
struct ETDesc { int sI, dI, aI, E, sT, dT, p0, p1; bool we; };

extern "C" void kernel_launch(void* const* d_in, const int* in_sizes, int n_in,
                              void* d_out, int out_size, void* d_ws, size_t ws_size,
                              hipStream_t stream) {
  (void)in_sizes; (void)n_in; (void)out_size; (void)ws_size;
  auto F = [&](int i) -> const float* { return (const float*)d_in[i]; };
  auto I = [&](int i) -> const int*   { return (const int*)d_in[i]; };

  static const int Nt[4] = {100000, 5000, 50000, 10000};
  static const int K0[4] = {16, 8, 12, 8};
  static const int XI[4] = {63, 64, 65, 66};
  static const int BI[4] = {67, 68, 69, 70};
  size_t obase[4];
  obase[0] = 0;
  obase[1] = (size_t)100000 * 256;
  obase[2] = (size_t)105000 * 256;
  obase[3] = (size_t)155000 * 256;

  // edge types: pre, proc, bcr (dst opes); procr (dst mas); bc, bjr (dst combs); bj (dst jobs)
  static const ETDesc ets[7] = {
    {71, 72, -1, 150000, 0, 0, 19, 49, false},  // pre   opes->opes
    {73, 74, 75, 300000, 1, 0, 23, 53, true },  // proc  mas->opes (We)
    {81, 82, -1, 150000, 2, 0,  7, 37, false},  // bcr   combs->opes
    {76, 77, 78, 300000, 0, 1, 28, 58, true },  // procr opes->mas (We)
    {79, 80, -1, 150000, 0, 2,  3, 33, false},  // bc    opes->combs
    {85, 86, -1,  80000, 3, 2, 15, 45, false},  // bjr   jobs->combs
    {83, 84, -1,  80000, 2, 3, 11, 41, false},  // bj    combs->jobs
  };

  // ---- workspace carve (256B aligned) ----
  char* wp = (char*)d_ws;
  auto carve = [&](size_t bytes) -> void* {
    void* p = (void*)wp;
    wp += (bytes + 255) & ~(size_t)255;
    return p;
  };
  float* o[4];
  for (int t = 0; t < 4; ++t) o[t] = (float*)carve((size_t)Nt[t] * 128 * 4);
  float*        xl    = (float*)carve((size_t)100000 * 128 * 4);
  float*        xr    = (float*)carve((size_t)100000 * 128 * 4);
  float*        alpha = (float*)carve((size_t)300000 * 8 * 4);
  unsigned int* amax  = (unsigned int*)carve((size_t)100000 * 8 * 4);
  float*        denom = (float*)carve((size_t)100000 * 8 * 4);
  float*        mean  = (float*)carve((size_t)NGRAPH * 128 * 4);
  float*        var   = (float*)carve((size_t)NGRAPH * 128 * 4);
  int* cnt[4];
  for (int t = 0; t < 4; ++t) cnt[t] = (int*)carve(NGRAPH * 4);

  auto g1 = [](int n) { return dim3((unsigned)((n + 255) / 256)); };

  // ---- per-graph node counts (shared across layers) ----
  for (int t = 0; t < 4; ++t) {
    hipMemsetAsync(cnt[t], 0, NGRAPH * 4, stream);
    k_count<<<g1(Nt[t]), 256, 0, stream>>>(I(BI[t]), cnt[t], Nt[t]);
  }

  float* outF = (float*)d_out;

  for (int L = 0; L < 2; ++L) {
    auto pbase = [&](const ETDesc& t) { return L == 0 ? t.p0 : t.p1; };
    auto bidx  = [&](const ETDesc& t) { return pbase(t) + (t.we ? 4 : 3); };

    // HeteroConv accumulators initialized with summed biases of incoming types
    k_bias<<<g1(Nt[0] * 128), 256, 0, stream>>>(o[0], F(bidx(ets[0])), F(bidx(ets[1])), F(bidx(ets[2])), Nt[0] * 128);
    k_bias<<<g1(Nt[1] * 128), 256, 0, stream>>>(o[1], F(bidx(ets[3])), nullptr, nullptr, Nt[1] * 128);
    k_bias<<<g1(Nt[2] * 128), 256, 0, stream>>>(o[2], F(bidx(ets[4])), F(bidx(ets[5])), nullptr, Nt[2] * 128);
    k_bias<<<g1(Nt[3] * 128), 256, 0, stream>>>(o[3], F(bidx(ets[6])), nullptr, nullptr, Nt[3] * 128);

    for (int q = 0; q < 7; ++q) {
      const ETDesc& t = ets[q];
      int base = pbase(t);
      const float* Wl  = F(base + (t.we ? 1 : 0));
      const float* Wr  = F(base + (t.we ? 2 : 1));
      const float* att = F(base + (t.we ? 3 : 2));
      const float* We  = t.we ? F(base) : nullptr;
      const float* ea  = (t.aI >= 0) ? F(t.aI) : nullptr;

      // xl = src features @ Wl ; xr = dst features @ Wr   (WMMA GEMMs)
      const float* As; int ldaS, Ks;
      if (L == 0) { As = F(XI[t.sT]); ldaS = Ks = K0[t.sT]; }
      else        { As = outF + obase[t.sT]; ldaS = 256; Ks = 128; }
      k_gemm_wmma<<<dim3((unsigned)((Nt[t.sT] + 15) / 16)), 256, 0, stream>>>(As, ldaS, Ks, Wl, xl, Nt[t.sT]);

      const float* Ad; int ldaD, Kd;
      if (L == 0) { Ad = F(XI[t.dT]); ldaD = Kd = K0[t.dT]; }
      else        { Ad = outF + obase[t.dT]; ldaD = 256; Kd = 128; }
      k_gemm_wmma<<<dim3((unsigned)((Nt[t.dT] + 15) / 16)), 256, 0, stream>>>(Ad, ldaD, Kd, Wr, xr, Nt[t.dT]);

      hipMemsetAsync(amax,  0, (size_t)Nt[t.dT] * 8 * 4, stream);
      hipMemsetAsync(denom, 0, (size_t)Nt[t.dT] * 8 * 4, stream);

      int EH = t.E * 8;
      k_edge_alpha<<<g1(EH), 256, 0, stream>>>(xl, xr, I(t.sI), I(t.dI), ea, We, att, alpha, amax, t.E);
      k_edge_exp  <<<g1(EH), 256, 0, stream>>>(alpha, I(t.dI), amax, denom, t.E);
      k_edge_aggr <<<g1(t.E * 32), 256, 0, stream>>>(xl, I(t.sI), I(t.dI), alpha, denom, o[t.dT], t.E);
    }

    // leaky(0.01) + GraphNorm per node type; write normalized features into d_out
    // (layer-1 GEMMs read layer-0 half of d_out directly, lda=256)
    for (int t = 0; t < 4; ++t) {
      int n32 = Nt[t] * 32;
      hipMemsetAsync(mean, 0, (size_t)NGRAPH * 128 * 4, stream);
      hipMemsetAsync(var,  0, (size_t)NGRAPH * 128 * 4, stream);
      k_act_sum   <<<g1(n32), 256, 0, stream>>>(o[t], I(BI[t]), mean, Nt[t]);
      k_div_cnt   <<<g1(NGRAPH * 128), 256, 0, stream>>>(mean, cnt[t]);
      k_center_var<<<g1(n32), 256, 0, stream>>>(o[t], I(BI[t]), mean, F(1) /*gn.ms*/, var, Nt[t]);
      k_var_inv   <<<g1(NGRAPH * 128), 256, 0, stream>>>(var, cnt[t]);
      k_norm_final<<<g1(n32), 256, 0, stream>>>(o[t], I(BI[t]), var, F(2) /*gn.w*/, F(0) /*gn.b*/,
                                                outF + obase[t] + (size_t)L * 128, Nt[t]);
    }
  }
}